// SerializedAttention_43843026157911
// MI455X (gfx1250) — compile-verified
//
#include <hip/hip_runtime.h>
#include <cstddef>
#include <cstdint>

typedef __attribute__((ext_vector_type(2))) float v2f;
typedef __attribute__((ext_vector_type(8))) float v8f;

#define PATCH 128
#define CCH   128
#define NHEAD 8
#define HDIM  16
#define QS    132                 // padded row stride (floats) -> conflict-free column reads
#define TILE_F (PATCH*QS)         // 16896 floats per 128x128 fp32 tile
#define PBS   33                  // padded P-stage stride
#define SMEM_FLOATS (512 + 792 + 4*TILE_F + 8*16*PBS)   // 73112 floats = 292448 B

static __device__ __forceinline__ v8f wmma4(v2f a, v2f b, v8f c) {
    // D = A(16x4) * B(4x16) + C(16x16), fp32
    return __builtin_amdgcn_wmma_f32_16x16x4_f32(false, a, false, b, (short)0, c,
                                                 false, false);
}

__global__ __launch_bounds__(256, 1)
void serattn_fused(const float* __restrict__ feat,
                   const float* __restrict__ Wqkv,
                   const float* __restrict__ bqkv,
                   const float* __restrict__ Wproj,
                   const float* __restrict__ bproj,
                   const float* __restrict__ rpe,
                   const int*  __restrict__ gcoord,
                   const int*  __restrict__ order,
                   float* __restrict__ out)
{
    extern __shared__ float sm[];
    int*   ordL = (int*)sm;               // 128 ints
    int*   gcL  = ordL + PATCH;           // 384 ints
    float* rpeL = sm + 512;               // 99*8 floats
    float* XO   = sm + 1304;              // X (phase1) then O (phase2/3)
    float* Qb   = XO + TILE_F;
    float* Kb   = Qb + TILE_F;
    float* Vb   = Kb + TILE_F;
    float* Pall = Vb + TILE_F;            // 8 waves x 16 x 33

    const int p    = blockIdx.x;
    const int tid  = threadIdx.x;
    const int wave = tid >> 5;
    const int lane = tid & 31;
    const int lo   = lane & 15;           // column-in-tile / row-in-A
    const int hi   = lane >> 4;           // lane-half
    const int kb   = hi * 2;              // K base for A/B fragments

    // ---------------- Phase 0: gather patch data into LDS ----------------
    for (int i = tid; i < PATCH; i += 256) ordL[i] = order[p*PATCH + i];
    __syncthreads();
    for (int i = tid; i < PATCH; i += 256) {
        int t = ordL[i];
        gcL[i*3+0] = gcoord[t*3+0];
        gcL[i*3+1] = gcoord[t*3+1];
        gcL[i*3+2] = gcoord[t*3+2];
    }
    for (int i = tid; i < 99*NHEAD; i += 256) rpeL[i] = rpe[i];

    // Async DMA of the 128 gathered feat rows straight into LDS.
    // One GLOBAL_LOAD_ASYNC_TO_LDS_B128 per row per wave: 32 lanes x 16B = 512B row.
    for (int row = wave; row < PATCH; row += 8) {
        const char* src = (const char*)(feat + (size_t)ordL[row]*CCH) + lane*16;
        unsigned    dst = (unsigned)(uintptr_t)(XO + row*QS) + (unsigned)(lane*16);
        asm volatile("global_load_async_to_lds_b128 %0, %1, off"
                     :: "v"(dst), "v"(src) : "memory");
    }
    asm volatile("s_wait_asynccnt 0x0" ::: "memory");
    __syncthreads();

    // ---------------- Phase 1: QKV = Xg * Wqkv + bqkv ----------------
    // 8 mtiles x 24 ntiles = 192 output tiles, 24 per wave
    for (int t = wave; t < 192; t += 8) {
        int mt = t / 24, nt = t - mt*24;
        int m0 = mt*16, n0 = nt*16;
        int col = n0 + lo;                     // 0..383
        float bq = bqkv[col];
        v8f acc = {bq,bq,bq,bq,bq,bq,bq,bq};
        const float* xrow = &XO[(m0 + lo)*QS];
        #pragma unroll 4
        for (int k0 = 0; k0 < CCH; k0 += 4) {
            v2f a = { xrow[k0 + kb], xrow[k0 + kb + 1] };
            const float* wp = Wqkv + (size_t)(k0 + kb)*384 + col;
            __builtin_prefetch(wp + 16*384, 0, 1);   // global_prefetch_b8
            v2f b = { wp[0], wp[384] };
            acc = wmma4(a, b, acc);
        }
        int s3 = col >> 7;                     // 0=Q 1=K 2=V (tile never crosses)
        int cc = col & 127;                    // h*16 + d
        float* dst = Qb + s3*TILE_F;
        #pragma unroll
        for (int r = 0; r < 8; r++)
            dst[(m0 + r + hi*8)*QS + cc] = acc[r];
    }
    __syncthreads();

    // ---------------- Phase 2: attention per (head, row-tile) ----------------
    float* pb = Pall + wave*(16*PBS);
    for (int t = wave; t < 64; t += 8) {
        int h  = t >> 3, mt = t & 7;
        int m0 = mt*16, hc = h*16;

        // S = Q_h * K_h^T  (8 column tiles held in registers)
        v8f S[8];
        #pragma unroll
        for (int nt = 0; nt < 8; nt++) {
            v8f acc = {};
            #pragma unroll
            for (int k0 = 0; k0 < HDIM; k0 += 4) {
                v2f a = { Qb[(m0 + lo)*QS + hc + k0 + kb],
                          Qb[(m0 + lo)*QS + hc + k0 + kb + 1] };
                v2f b = { Kb[(nt*16 + lo)*QS + hc + k0 + kb],
                          Kb[(nt*16 + lo)*QS + hc + k0 + kb + 1] };
                acc = wmma4(a, b, acc);
            }
            S[nt] = acc;
        }

        // scale + RPE bias + row softmax (row = m0 + r + 8*hi, col = nt*16 + lo)
        #pragma unroll
        for (int r = 0; r < 8; r++) {
            int i = m0 + r + hi*8;
            int gx = gcL[i*3+0], gy = gcL[i*3+1], gz = gcL[i*3+2];
            float mx = -1e30f;
            #pragma unroll
            for (int nt = 0; nt < 8; nt++) {
                int j  = nt*16 + lo;
                int dx = min(max(gx - gcL[j*3+0], -16), 16) + 16;        // a=0
                int dy = min(max(gy - gcL[j*3+1], -16), 16) + 16 + 33;   // a=1
                int dz = min(max(gz - gcL[j*3+2], -16), 16) + 16 + 66;   // a=2
                float v = S[nt][r]*0.25f
                        + rpeL[dx*NHEAD + h] + rpeL[dy*NHEAD + h] + rpeL[dz*NHEAD + h];
                S[nt][r] = v;
                mx = fmaxf(mx, v);
            }
            #pragma unroll
            for (int msk = 1; msk <= 8; msk <<= 1)
                mx = fmaxf(mx, __shfl_xor(mx, msk, 32));
            float sum = 0.f;
            #pragma unroll
            for (int nt = 0; nt < 8; nt++) {
                float e = __expf(S[nt][r] - mx);
                S[nt][r] = e; sum += e;
            }
            #pragma unroll
            for (int msk = 1; msk <= 8; msk <<= 1)
                sum += __shfl_xor(sum, msk, 32);
            float inv = 1.0f / sum;
            #pragma unroll
            for (int nt = 0; nt < 8; nt++) S[nt][r] *= inv;
        }

        // O = P * V_h : stage P in 32-column chunks through padded LDS (A-layout)
        v8f o = {};
        #pragma unroll
        for (int kc = 0; kc < 4; kc++) {
            #pragma unroll
            for (int half = 0; half < 2; half++) {
                int nt = kc*2 + half;
                #pragma unroll
                for (int r = 0; r < 8; r++)
                    pb[(r + hi*8)*PBS + half*16 + lo] = S[nt][r];
            }
            #pragma unroll
            for (int kk = 0; kk < 32; kk += 4) {
                v2f a = { pb[lo*PBS + kk + kb], pb[lo*PBS + kk + kb + 1] };
                int kt = kc*32 + kk + kb;
                v2f b = { Vb[kt*QS + hc + lo], Vb[(kt+1)*QS + hc + lo] };
                o = wmma4(a, b, o);
            }
        }
        #pragma unroll
        for (int r = 0; r < 8; r++)
            XO[(m0 + r + hi*8)*QS + hc + lo] = o[r];   // Xg region now holds O
    }
    __syncthreads();

    // ---------------- Phase 3: out = O * Wproj + bproj, scatter by order ----------------
    for (int t = wave; t < 64; t += 8) {
        int mt = t >> 3, nt = t & 7;
        int m0 = mt*16, n0 = nt*16;
        float bp = bproj[n0 + lo];
        v8f acc = {bp,bp,bp,bp,bp,bp,bp,bp};
        #pragma unroll 4
        for (int k0 = 0; k0 < CCH; k0 += 4) {
            v2f a = { XO[(m0 + lo)*QS + k0 + kb], XO[(m0 + lo)*QS + k0 + kb + 1] };
            const float* wp = Wproj + (size_t)(k0 + kb)*CCH + n0 + lo;
            __builtin_prefetch(wp + 16*CCH, 0, 1);
            v2f b = { wp[0], wp[CCH] };
            acc = wmma4(a, b, acc);
        }
        #pragma unroll
        for (int r = 0; r < 8; r++) {
            int row = m0 + r + hi*8;
            out[(size_t)ordL[row]*CCH + n0 + lo] = acc[r];   // == out[serialized_inverse]
        }
    }
}

extern "C" void kernel_launch(void* const* d_in, const int* in_sizes, int n_in,
                              void* d_out, int out_size, void* d_ws, size_t ws_size,
                              hipStream_t stream) {
    const float* feat  = (const float*)d_in[0];
    const float* Wqkv  = (const float*)d_in[1];
    const float* bqkv  = (const float*)d_in[2];
    const float* Wproj = (const float*)d_in[3];
    const float* bproj = (const float*)d_in[4];
    const float* rpe   = (const float*)d_in[5];
    const int*   gc    = (const int*)d_in[6];
    const int*   order = (const int*)d_in[7];
    float* out = (float*)d_out;

    int N  = in_sizes[0] / CCH;     // 131072
    int nP = N / PATCH;             // 1024 patches, one workgroup each
    size_t smem = (size_t)SMEM_FLOATS * sizeof(float);   // ~286 KB of 320 KB/WGP
    serattn_fused<<<nP, 256, smem, stream>>>(feat, Wqkv, bqkv, Wproj, bproj,
                                             rpe, gc, order, out);
}